// NSF_SAR_9698036154973
// MI455X (gfx1250) — compile-verified
//
#include <hip/hip_runtime.h>
#include <math.h>

typedef float v2f __attribute__((ext_vector_type(2)));
typedef float v8f __attribute__((ext_vector_type(8)));

#define DIMS   64
#define HID    64
#define NBATCH 16384
#define BTILE  128
#define PADW   68
#define PPAD   25
#define TAILB  3.0f

__device__ __forceinline__ float softplus_f(float x) {
  return fmaxf(x, 0.0f) + log1pf(expf(-fabsf(x)));
}

// p[0..7] -> bin edges / widths following the reference's double-softmax:
// u = 2*B_TAIL*softmax(p); w = softmax(u); width = 1e-3 + (1-8e-3)*w; edges = 6*cumsum - 3
__device__ __forceinline__ void build_edges(const float* p, float* edges, float* widths) {
  float m1 = p[0];
#pragma unroll
  for (int j = 1; j < 8; ++j) m1 = fmaxf(m1, p[j]);
  float e[8];
  float s = 0.0f;
#pragma unroll
  for (int j = 0; j < 8; ++j) { e[j] = expf(p[j] - m1); s += e[j]; }
  float inv = (2.0f * TAILB) / s;  // u_j = 6 * softmax_j
  float u[8];
  float m2 = -1e30f;
#pragma unroll
  for (int j = 0; j < 8; ++j) { u[j] = e[j] * inv; m2 = fmaxf(m2, u[j]); }
  float s2 = 0.0f;
#pragma unroll
  for (int j = 0; j < 8; ++j) { e[j] = expf(u[j] - m2); s2 += e[j]; }
  float inv2 = (1.0f - 8.0f * 1e-3f) / s2;
  float cum = 0.0f;
  edges[0] = -TAILB;
#pragma unroll
  for (int j = 0; j < 8; ++j) {
    float w = 1e-3f + e[j] * inv2;
    cum += w;
    edges[j + 1] = 2.0f * TAILB * cum - TAILB;
  }
  edges[8] = TAILB;
#pragma unroll
  for (int j = 0; j < 8; ++j) widths[j] = edges[j + 1] - edges[j];
}

__device__ __forceinline__ void rqs_spline(const float* p, float xin, float& zout, float& ldout) {
  bool inside = (xin >= -TAILB) && (xin <= TAILB);
  float xc = fminf(fmaxf(xin, -TAILB), TAILB);
  float cw[9], wdt[8], ch[9], hgt[8], der[9];
  build_edges(p, cw, wdt);
  build_edges(p + 8, ch, hgt);
  der[0] = 1.0f;  // 1e-3 + softplus(log(exp(1-1e-3)-1)) == 1 exactly
  der[8] = 1.0f;
#pragma unroll
  for (int j = 1; j <= 7; ++j) der[j] = 1e-3f + softplus_f(softplus_f(p[15 + j]));
  int idx = -1;
#pragma unroll
  for (int j = 0; j < 9; ++j) {
    float edge = cw[j] + ((j == 8) ? 1e-6f : 0.0f);
    idx += (xc >= edge) ? 1 : 0;
  }
  idx = max(0, min(idx, 7));
  float icw = cw[idx], iw = wdt[idx];
  float ich = ch[idx], ih = hgt[idx];
  float delta = ih / iw;
  float dk = der[idx], dk1 = der[idx + 1];
  float th  = (xc - icw) / iw;
  float omt = 1.0f - th;
  float t1  = th * omt;
  float numer = ih * (delta * th * th + dk * t1);
  float denom = delta + (dk + dk1 - 2.0f * delta) * t1;
  float outv  = ich + numer / denom;
  float dnum  = delta * delta * (dk1 * th * th + 2.0f * delta * t1 + dk * omt * omt);
  float ld    = logf(dnum) - 2.0f * logf(denom);
  zout  = inside ? outv : xin;
  ldout = inside ? ld : 0.0f;
}

__global__ void __launch_bounds__(256)
nsf_main(const float* __restrict__ x,  const float* __restrict__ initp,
         const float* __restrict__ W1, const float* __restrict__ b1,
         const float* __restrict__ W2, const float* __restrict__ b2,
         const float* __restrict__ W3, const float* __restrict__ b3,
         float* __restrict__ z_out, float* __restrict__ ld_ws) {
  __shared__ float Xs[BTILE][PADW];
  __shared__ float Hs[BTILE][PADW];
  __shared__ float W1s[HID][PADW];
  __shared__ float W2s[HID][PADW];
  __shared__ float W3s[32][PADW];
  __shared__ float B1s[HID];
  __shared__ float B2s[HID];
  __shared__ float B3s[32];
  __shared__ float Ps[BTILE][PPAD];

  const int d   = blockIdx.y;
  const int b0  = blockIdx.x * BTILE;
  const int tid = threadIdx.x;

  if (d == 0) {
    // dimension 0: broadcast init_param spline, no MLP
    if (tid < BTILE) {
      int b = b0 + tid;
      float pl[23];
#pragma unroll
      for (int j = 0; j < 23; ++j) pl[j] = initp[j];
      float z, ld;
      rqs_spline(pl, x[b * DIMS + 0], z, ld);
      z_out[b * DIMS + 0] = z;
      ld_ws[(size_t)0 * NBATCH + b] = ld;
    }
    return;
  }

  const int l = d - 1;

  // --- cooperative stage: masked X tile + layer weights into LDS ---
  for (int i = tid; i < BTILE * DIMS; i += 256) {
    int m = i >> 6, j = i & 63;
    float v = x[(b0 + m) * DIMS + j];
    Xs[m][j] = (j < d) ? v : 0.0f;  // autoregressive mask: keep columns 0..d-1
  }
  {
    const float* w1p = W1 + (size_t)l * HID * DIMS;
    const float* w2p = W2 + (size_t)l * HID * HID;
    for (int i = tid; i < HID * DIMS; i += 256) {
      W1s[i >> 6][i & 63] = w1p[i];
      W2s[i >> 6][i & 63] = w2p[i];
    }
    const float* w3p = W3 + (size_t)l * 23 * HID;
    for (int i = tid; i < 32 * HID; i += 256) {
      int r = i >> 6, c = i & 63;
      W3s[r][c] = (r < 23) ? w3p[r * HID + c] : 0.0f;
    }
    if (tid < 64) {
      B1s[tid] = b1[l * HID + tid];
      B2s[tid] = b2[l * HID + tid];
    } else if (tid < 96) {
      int r = tid - 64;
      B3s[r] = (r < 23) ? b3[l * 23 + r] : 0.0f;
    }
  }
  __syncthreads();

  const int lane = tid & 31;
  const int wave = tid >> 5;
  const int mrow = wave * 16;          // this wave's 16-row M slab
  const int ln   = lane & 15;          // N (or M for A-operand) within tile
  const int kh   = (lane >> 4) << 1;   // A/B half: K offset 0 or 2
  const int mh   = (lane >> 4) << 3;   // D half: row offset 0 or 8

  const int kmax1 = (d + 3) & ~3;      // masked columns beyond d are zero

  // --- Layer 1: Hs = tanh(Xs @ W1^T + b1) ---
#pragma unroll
  for (int nt = 0; nt < 4; ++nt) {
    const int n0 = nt * 16;
    const float bias = B1s[n0 + ln];
    v8f acc;
#pragma unroll
    for (int r = 0; r < 8; ++r) acc[r] = bias;
    for (int k0 = 0; k0 < kmax1; k0 += 4) {
      v2f a, bb;
      a.x  = Xs[mrow + ln][k0 + kh];
      a.y  = Xs[mrow + ln][k0 + kh + 1];
      bb.x = W1s[n0 + ln][k0 + kh];
      bb.y = W1s[n0 + ln][k0 + kh + 1];
      acc = __builtin_amdgcn_wmma_f32_16x16x4_f32(false, a, false, bb, (short)0, acc,
                                                  false, false);
    }
#pragma unroll
    for (int r = 0; r < 8; ++r) Hs[mrow + mh + r][n0 + ln] = tanhf(acc[r]);
  }
  __syncthreads();

  // --- Layer 2: Xs = tanh(Hs @ W2^T + b2)  (reuse Xs for h2) ---
#pragma unroll
  for (int nt = 0; nt < 4; ++nt) {
    const int n0 = nt * 16;
    const float bias = B2s[n0 + ln];
    v8f acc;
#pragma unroll
    for (int r = 0; r < 8; ++r) acc[r] = bias;
    for (int k0 = 0; k0 < HID; k0 += 4) {
      v2f a, bb;
      a.x  = Hs[mrow + ln][k0 + kh];
      a.y  = Hs[mrow + ln][k0 + kh + 1];
      bb.x = W2s[n0 + ln][k0 + kh];
      bb.y = W2s[n0 + ln][k0 + kh + 1];
      acc = __builtin_amdgcn_wmma_f32_16x16x4_f32(false, a, false, bb, (short)0, acc,
                                                  false, false);
    }
#pragma unroll
    for (int r = 0; r < 8; ++r) Xs[mrow + mh + r][n0 + ln] = tanhf(acc[r]);
  }
  __syncthreads();

  // --- Layer 3: Ps = h2 @ W3^T + b3 (23 cols, padded) ---
#pragma unroll
  for (int nt = 0; nt < 2; ++nt) {
    const int n0 = nt * 16;
    const float bias = B3s[n0 + ln];
    v8f acc;
#pragma unroll
    for (int r = 0; r < 8; ++r) acc[r] = bias;
    for (int k0 = 0; k0 < HID; k0 += 4) {
      v2f a, bb;
      a.x  = Xs[mrow + ln][k0 + kh];
      a.y  = Xs[mrow + ln][k0 + kh + 1];
      bb.x = W3s[n0 + ln][k0 + kh];
      bb.y = W3s[n0 + ln][k0 + kh + 1];
      acc = __builtin_amdgcn_wmma_f32_16x16x4_f32(false, a, false, bb, (short)0, acc,
                                                  false, false);
    }
    const int col = n0 + ln;
#pragma unroll
    for (int r = 0; r < 8; ++r)
      if (col < 23) Ps[mrow + mh + r][col] = acc[r];
  }
  __syncthreads();

  // --- Spline: one thread per batch row of the tile ---
  if (tid < BTILE) {
    int b = b0 + tid;
    float z, ld;
    rqs_spline(&Ps[tid][0], x[b * DIMS + d], z, ld);
    z_out[b * DIMS + d] = z;
    ld_ws[(size_t)d * NBATCH + b] = ld;
  }
}

__global__ void __launch_bounds__(256)
nsf_reduce(const float* __restrict__ ld_ws, float* __restrict__ out_ld) {
  int b = blockIdx.x * 256 + threadIdx.x;
  float s = 0.0f;
#pragma unroll 8
  for (int dd = 0; dd < DIMS; ++dd) s += ld_ws[(size_t)dd * NBATCH + b];
  out_ld[b] = s;
}

extern "C" void kernel_launch(void* const* d_in, const int* in_sizes, int n_in,
                              void* d_out, int out_size, void* d_ws, size_t ws_size,
                              hipStream_t stream) {
  const float* x     = (const float*)d_in[0];
  const float* initp = (const float*)d_in[1];
  const float* W1    = (const float*)d_in[2];
  const float* b1    = (const float*)d_in[3];
  const float* W2    = (const float*)d_in[4];
  const float* b2    = (const float*)d_in[5];
  const float* W3    = (const float*)d_in[6];
  const float* b3    = (const float*)d_in[7];

  float* z_out  = (float*)d_out;                       // [NBATCH, DIMS]
  float* ld_out = z_out + (size_t)NBATCH * DIMS;       // [NBATCH]
  float* ld_ws  = (float*)d_ws;                        // [DIMS, NBATCH] scratch (4 MB)

  dim3 grid(NBATCH / BTILE, DIMS);                     // 128 x 64 blocks
  nsf_main<<<grid, 256, 0, stream>>>(x, initp, W1, b1, W2, b2, W3, b3, z_out, ld_ws);
  nsf_reduce<<<NBATCH / 256, 256, 0, stream>>>(ld_ws, ld_out);
}